// ConstituentAttention_81853486727892
// MI455X (gfx1250) — compile-verified
//
#include <hip/hip_runtime.h>
#include <hip/hip_bf16.h>
#include <math.h>

typedef __attribute__((ext_vector_type(2))) float v2f;
typedef __attribute__((ext_vector_type(8))) float v8f;

#define EMBED 1024
#define PROJ  128
#define NOUT  256
#define BATCH 32
#define SEQ   1024
#define M_TOT (BATCH * SEQ)

// ---------------------------------------------------------------------------
// Kernel A: qk[m, n] = sum_e ctx[m, e] * W[n, e] + bias[n]
// ctx row m=(b*S+s) lives at context + (s*B + b)*E (context is [S,B,E]).
// Block: 256 thr (8 waves). Block tile = 32 M rows x 256 N cols.
// Each wave: 2 M-tiles x 2 N-tiles (32x32 area, 4 v8f accumulators) so each
// A and B fragment load feeds two v_wmma_f32_16x16x4_f32 ops (1 DS load/WMMA).
// ---------------------------------------------------------------------------
__global__ __launch_bounds__(256) void gemm_qk_kernel(
    const float* __restrict__ ctx, const float* __restrict__ W,
    const float* __restrict__ bias, float* __restrict__ qk) {
  constexpr int KC = 32;    // K chunk
  constexpr int KCP = 36;   // padded LDS stride (stride 36 mod 64 banks: clean)
  __shared__ float As[32 * KCP];     // 4.6 KB
  __shared__ float Ws[NOUT * KCP];   // 36.9 KB

  const int t    = threadIdx.x;
  const int wave = t >> 5;
  const int lane = t & 31;
  const int half = lane >> 4;   // 0: K 0,1   1: K 2,3
  const int l    = lane & 15;
  const int nbase = wave << 5;  // 32-wide N strip per wave

  const int m0 = blockIdx.x << 5;         // first of 32 rows
  const int b  = m0 >> 10;                // SEQ = 1024
  const int s0 = m0 & (SEQ - 1);          // 32-row tile never crosses batch
  const float* arow0 = ctx + ((size_t)s0 * BATCH + b) * EMBED;

  v8f c00 = {}, c01 = {}, c10 = {}, c11 = {};

  // A staging: 32 rows x 32 k = 1024 floats -> exactly 1 float4 per thread.
  const int ar  = t >> 3;       // 0..31
  const int ak4 = t & 7;        // 0..7
  const float* aptr = arow0 + (size_t)ar * BATCH * EMBED + ak4 * 4;

  for (int k0 = 0; k0 < EMBED; k0 += KC) {
    // Issue all global loads first (registers), then all LDS stores, so the
    // compiler can clause the loads behind one s_wait_loadcnt.
    float4 areg = *(const float4*)(aptr + k0);
    float4 wreg[8];
#pragma unroll
    for (int j = 0; j < 8; ++j) {
      int g = j * 256 + t;   // 0..2047 float4s of the 256x32 W tile
      int n = g >> 3;
      int k4 = g & 7;
      wreg[j] = *(const float4*)(W + (size_t)n * EMBED + k0 + k4 * 4);
    }
    *(float4*)&As[ar * KCP + ak4 * 4] = areg;
#pragma unroll
    for (int j = 0; j < 8; ++j) {
      int g = j * 256 + t;
      int n = g >> 3;
      int k4 = g & 7;
      *(float4*)&Ws[n * KCP + k4 * 4] = wreg[j];
    }
    __syncthreads();

#pragma unroll
    for (int kk = 0; kk < KC; kk += 4) {
      const int kA = kk + half * 2;
      // A 16x4 fragments (two M tiles): lanes 0-15 K kA,kA+1; lanes 16-31 +2.
      v2f a0, a1;
      a0.x = As[l * KCP + kA];
      a0.y = As[l * KCP + kA + 1];
      a1.x = As[(16 + l) * KCP + kA];
      a1.y = As[(16 + l) * KCP + kA + 1];
      // B 4x16 fragments (two N tiles); B[k][n] = W[n][k], mirror layout.
      v2f b0, b1;
      b0.x = Ws[(nbase + l) * KCP + kA];
      b0.y = Ws[(nbase + l) * KCP + kA + 1];
      b1.x = Ws[(nbase + 16 + l) * KCP + kA];
      b1.y = Ws[(nbase + 16 + l) * KCP + kA + 1];
      c00 = __builtin_amdgcn_wmma_f32_16x16x4_f32(false, a0, false, b0,
                                                  (short)0, c00, false, false);
      c01 = __builtin_amdgcn_wmma_f32_16x16x4_f32(false, a0, false, b1,
                                                  (short)0, c01, false, false);
      c10 = __builtin_amdgcn_wmma_f32_16x16x4_f32(false, a1, false, b0,
                                                  (short)0, c10, false, false);
      c11 = __builtin_amdgcn_wmma_f32_16x16x4_f32(false, a1, false, b1,
                                                  (short)0, c11, false, false);
    }
    __syncthreads();
  }

  // C/D layout: VGPR i -> row i (lanes 0-15) / row i+8 (lanes 16-31), col=lane%16
  const float bias0 = bias[nbase + l];
  const float bias1 = bias[nbase + 16 + l];
#pragma unroll
  for (int i = 0; i < 8; ++i) {
    int r = i + half * 8;
    size_t row_a = (size_t)(m0 + r) * NOUT;        // M tile 0
    size_t row_b = (size_t)(m0 + 16 + r) * NOUT;   // M tile 1
    qk[row_a + nbase + l]      = c00[i] + bias0;
    qk[row_a + nbase + 16 + l] = c01[i] + bias1;
    qk[row_b + nbase + l]      = c10[i] + bias0;
    qk[row_b + nbase + 16 + l] = c11[i] + bias1;
  }
}

// ---------------------------------------------------------------------------
// Kernel B: fwd[b,i] = q[b,i].k[b,i+1]/E ; bwd[b,i] = q[b,i+1].k[b,i]/E
// One wave per position i (i < S-1). 128-dim dot: one float4 per lane.
// ---------------------------------------------------------------------------
__global__ __launch_bounds__(256) void neighbor_dots_kernel(
    const float* __restrict__ qk, float* __restrict__ fwd,
    float* __restrict__ bwd) {
  const int wglobal = blockIdx.x * 8 + (threadIdx.x >> 5);
  const int lane = threadIdx.x & 31;
  const int total = BATCH * (SEQ - 1);
  if (wglobal >= total) return;
  const int b = wglobal / (SEQ - 1);
  const int i = wglobal % (SEQ - 1);
  const size_t ri = (size_t)(b * SEQ + i) * NOUT;
  const size_t rj = ri + NOUT;  // row i+1
  float4 qi = *(const float4*)(qk + ri + lane * 4);
  float4 kj = *(const float4*)(qk + rj + PROJ + lane * 4);
  float4 qj = *(const float4*)(qk + rj + lane * 4);
  float4 ki = *(const float4*)(qk + ri + PROJ + lane * 4);
  float f = qi.x * kj.x + qi.y * kj.y + qi.z * kj.z + qi.w * kj.w;
  float g = qj.x * ki.x + qj.y * ki.y + qj.z * ki.z + qj.w * ki.w;
  for (int off = 16; off > 0; off >>= 1) {
    f += __shfl_down(f, off, 32);
    g += __shfl_down(g, off, 32);
  }
  if (lane == 0) {
    fwd[b * SEQ + i] = f * (1.0f / (float)EMBED);
    bwd[b * SEQ + i] = g * (1.0f / (float)EMBED);
  }
}

// ---------------------------------------------------------------------------
// Kernel C1: 2-way softmax over stacked (col0, col1) with edge -inf + mask.
// ---------------------------------------------------------------------------
__global__ __launch_bounds__(256) void probs_kernel(
    const float* __restrict__ fwd, const float* __restrict__ bwd,
    const int* __restrict__ mask, float* __restrict__ prob0,
    float* __restrict__ prob1) {
  const int idx = blockIdx.x * 256 + threadIdx.x;
  if (idx >= M_TOT) return;
  const int b = idx >> 10;
  const int s = idx & (SEQ - 1);
  float c0 = (s < SEQ - 1) ? fwd[idx] : -INFINITY;
  const int rm = mask[b * SEQ + ((s + 1) & (SEQ - 1))];  // rolled mask
  if (rm == 1) c0 = -INFINITY;
  float c1 = (s > 0) ? bwd[idx - 1] : -INFINITY;
  const float mx = fmaxf(c0, c1);
  const float e0 = expf(c0 - mx);
  const float e1 = expf(c1 - mx);
  const float inv = 1.0f / (e0 + e1);
  prob0[idx] = e0 * inv;
  prob1[idx] = e1 * inv;
}

// ---------------------------------------------------------------------------
// Kernel C2: flat-rolled shift, neighbor_attn, log, exclusive prefix scan.
// One 1024-thread block per batch.
// ---------------------------------------------------------------------------
__global__ __launch_bounds__(1024) void scan_batch_kernel(
    const float* __restrict__ prob0, const float* __restrict__ prob1,
    const float* __restrict__ prior, const int* __restrict__ mask,
    float* __restrict__ neighbor_out, float* __restrict__ csp) {
  __shared__ float sh[SEQ];
  const int b = blockIdx.x;
  const int s = threadIdx.x;
  const int idx = b * SEQ + s;
  // torch-style flat roll(-1) crosses batch boundaries:
  const float sp = prob1[(idx + 1) % M_TOT];
  const float pv = sqrtf(prob0[idx] * sp + 1e-6f);
  const float pr = prior[idx];
  const float na = pr + (1.0f - pr) * pv;
  neighbor_out[idx] = na;
  float lp = logf(na);
  const int rm = mask[b * SEQ + ((s + 1) & (SEQ - 1))];
  if (rm == 1) lp = 0.0f;
  sh[s] = lp;
  __syncthreads();
  // Hillis-Steele inclusive scan over 1024 elements
  for (int off = 1; off < SEQ; off <<= 1) {
    float v = (s >= off) ? sh[s - off] : 0.0f;
    __syncthreads();
    sh[s] += v;
    __syncthreads();
  }
  csp[idx] = (s == 0) ? 0.0f : sh[s - 1];  // exclusive
}

// ---------------------------------------------------------------------------
// Kernel D: constituent_attn[b,i,j] = (i==j) ? 0 : exp(csp[max]-csp[min]).
// Pure streaming-store kernel (134 MB), csp rows cache-resident.
// ---------------------------------------------------------------------------
__global__ __launch_bounds__(256) void constituent_kernel(
    const float* __restrict__ csp, float* __restrict__ out) {
  const int j = blockIdx.x * 256 + threadIdx.x;
  const int i = blockIdx.y;
  const int b = blockIdx.z;
  const float ci = csp[b * SEQ + i];
  const float cj = csp[b * SEQ + j];
  const float d = (j > i) ? (cj - ci) : (ci - cj);
  const float r = (i == j) ? 0.0f : expf(d);
  out[(size_t)(b * SEQ + i) * SEQ + j] = r;
}

extern "C" void kernel_launch(void* const* d_in, const int* in_sizes, int n_in,
                              void* d_out, int out_size, void* d_ws,
                              size_t ws_size, hipStream_t stream) {
  const float* context = (const float*)d_in[0];  // [S, B, E]
  const float* prior   = (const float*)d_in[1];  // [B, S]
  const int*   mask    = (const int*)d_in[2];    // [B, S]
  const float* W       = (const float*)d_in[3];  // [256, E]
  const float* bias    = (const float*)d_in[4];  // [256]
  float* out = (float*)d_out;  // [B,S,S] constituent ++ [B,S] neighbor

  float* ws    = (float*)d_ws;
  float* qk    = ws;                          // 32768 * 256
  float* fwd   = qk + (size_t)M_TOT * NOUT;   // 32768
  float* bwd   = fwd + M_TOT;                 // 32768
  float* prob0 = bwd + M_TOT;                 // 32768
  float* prob1 = prob0 + M_TOT;               // 32768
  float* csp   = prob1 + M_TOT;               // 32768

  float* neighbor_out = out + (size_t)M_TOT * SEQ;

  gemm_qk_kernel<<<M_TOT / 32, 256, 0, stream>>>(context, W, bias, qk);

  const int nwaves = BATCH * (SEQ - 1);
  neighbor_dots_kernel<<<(nwaves + 7) / 8, 256, 0, stream>>>(qk, fwd, bwd);

  probs_kernel<<<(M_TOT + 255) / 256, 256, 0, stream>>>(fwd, bwd, mask, prob0,
                                                        prob1);

  scan_batch_kernel<<<BATCH, 1024, 0, stream>>>(prob0, prob1, prior, mask,
                                                neighbor_out, csp);

  constituent_kernel<<<dim3(SEQ / 256, SEQ, BATCH), 256, 0, stream>>>(csp, out);
}